// GCN4_31379031064900
// MI455X (gfx1250) — compile-verified
//
#include <hip/hip_runtime.h>
#include <math.h>

#define NNODES 50000
#define NEDGES 800000
#define FDIM   256

typedef float v2f __attribute__((ext_vector_type(2)));
typedef float v8f __attribute__((ext_vector_type(8)));

// ---------------- degree / norm ----------------

__global__ void deg_init_kernel(unsigned* __restrict__ deg, int n) {
    int i = blockIdx.x * blockDim.x + threadIdx.x;
    if (i < n) deg[i] = 1u;  // self-loop
}

__global__ void deg_edge_kernel(const long long* __restrict__ ei,
                                unsigned* __restrict__ deg, int e) {
    int t = blockIdx.x * blockDim.x + threadIdx.x;
    if (t < e) {
        int d = (int)ei[e + t];  // dst row of edge_index
        atomicAdd(&deg[d], 1u);
    }
}

__global__ void dis_kernel(const unsigned* __restrict__ deg,
                           float* __restrict__ dis, int n) {
    int i = blockIdx.x * blockDim.x + threadIdx.x;
    if (i < n) dis[i] = rsqrtf((float)deg[i]);
}

// ---------------- f32 WMMA GEMM: T = act(H) @ W  (H: N x 256, W: 256 x 256) ----------------
// One wave computes one 16x16 tile of T. K-loop of 64 steps of V_WMMA_F32_16X16X4_F32.
// Grid is sized so (#waves == #tiles) exactly -> EXEC is all-ones for every WMMA.

template <int RELU>
__global__ void gemm256_wmma_kernel(const float* __restrict__ H,
                                    const float* __restrict__ W,
                                    float* __restrict__ T) {
    const int wid  = (blockIdx.x * blockDim.x + threadIdx.x) >> 5;  // tile id, wave-uniform
    const int lane = threadIdx.x & 31;

    const int rowTile = wid >> 4;        // 0..3124
    const int colTile = wid & 15;        // 0..15

    const int l15  = lane & 15;
    const int half = lane >> 4;          // 0 or 1
    const int kofs = half * 2;           // K sub-offset within each K=4 step

    const float* __restrict__ arow = H + (size_t)(rowTile * 16 + l15) * FDIM;
    const int col = colTile * 16 + l15;

    v8f c = {};
#pragma unroll 4
    for (int k0 = 0; k0 < FDIM; k0 += 4) {
        // A tile 16x4: lane holds H[row, k0+kofs], H[row, k0+kofs+1]
        float2 av = *(const float2*)(arow + k0 + kofs);
        if (RELU) {
            av.x = fmaxf(av.x, 0.0f);
            av.y = fmaxf(av.y, 0.0f);
        }
        v2f a = {av.x, av.y};
        // B tile 4x16: lane holds W[k0+kofs, col], W[k0+kofs+1, col]
        v2f b = {W[(size_t)(k0 + kofs) * FDIM + col],
                 W[(size_t)(k0 + kofs + 1) * FDIM + col]};
        c = __builtin_amdgcn_wmma_f32_16x16x4_f32(false, a, false, b,
                                                  (short)0, c, false, false);
    }

    // C/D layout: VGPR v, lane L -> row v + 8*(L>>4), col L&15
    const int mBase = rowTile * 16 + half * 8;
#pragma unroll
    for (int v = 0; v < 8; ++v) {
        T[(size_t)(mBase + v) * FDIM + col] = c[v];
    }
}

// ---------------- aggregation ----------------
// Agg[i,j] = b[j] + dis[i]^2 * T[i,j]   (bias + self-loop message)
__global__ void agg_init_kernel(const float* __restrict__ T,
                                const float* __restrict__ dis,
                                const float* __restrict__ bias,
                                float* __restrict__ Agg, int total) {
    int idx = blockIdx.x * blockDim.x + threadIdx.x;
    if (idx < total) {
        int i = idx >> 8;
        int j = idx & 255;
        float di = dis[i];
        Agg[idx] = bias[j] + di * di * T[idx];
    }
}

// One edge handled by 64 threads, each owning a float4 feature chunk.
__global__ void agg_edge_kernel(const long long* __restrict__ ei,
                                const float* __restrict__ T,
                                const float* __restrict__ dis,
                                float* __restrict__ Agg, int e) {
    int t = blockIdx.x * blockDim.x + threadIdx.x;
    int edge = t >> 6;
    if (edge >= e) return;
    int c4 = t & 63;

    int s = (int)ei[edge];
    int d = (int)ei[e + edge];
    float nrm = dis[s] * dis[d];

    const float4 v = *(const float4*)(T + (size_t)s * FDIM + c4 * 4);
    float* o = Agg + (size_t)d * FDIM + c4 * 4;
    unsafeAtomicAdd(o + 0, v.x * nrm);
    unsafeAtomicAdd(o + 1, v.y * nrm);
    unsafeAtomicAdd(o + 2, v.z * nrm);
    unsafeAtomicAdd(o + 3, v.w * nrm);
}

// ---------------- final layer (256 -> 2) ----------------
// One wave per node row: T3[i, 0:2] = relu(Agg[i,:]) @ W3
__global__ void t3_kernel(const float* __restrict__ Agg,
                          const float* __restrict__ W3,
                          float* __restrict__ T3, int n) {
    int wid  = (blockIdx.x * blockDim.x + threadIdx.x) >> 5;
    int lane = threadIdx.x & 31;
    if (wid >= n) return;

    const float* h = Agg + (size_t)wid * FDIM;
    float s0 = 0.0f, s1 = 0.0f;
#pragma unroll
    for (int c = lane; c < FDIM; c += 32) {
        float v = fmaxf(h[c], 0.0f);
        s0 += v * W3[c * 2 + 0];
        s1 += v * W3[c * 2 + 1];
    }
#pragma unroll
    for (int off = 16; off > 0; off >>= 1) {
        s0 += __shfl_xor(s0, off, 32);
        s1 += __shfl_xor(s1, off, 32);
    }
    if (lane == 0) {
        T3[wid * 2 + 0] = s0;
        T3[wid * 2 + 1] = s1;
    }
}

__global__ void out_init_kernel(const float* __restrict__ T3,
                                const float* __restrict__ dis,
                                const float* __restrict__ b3,
                                float* __restrict__ out, int total) {
    int idx = blockIdx.x * blockDim.x + threadIdx.x;
    if (idx < total) {
        int i = idx >> 1;
        int k = idx & 1;
        float di = dis[i];
        out[idx] = b3[k] + di * di * T3[idx];
    }
}

__global__ void out_edge_kernel(const long long* __restrict__ ei,
                                const float* __restrict__ T3,
                                const float* __restrict__ dis,
                                float* __restrict__ out, int e) {
    int t = blockIdx.x * blockDim.x + threadIdx.x;
    int edge = t >> 1;
    if (edge >= e) return;
    int k = t & 1;

    int s = (int)ei[edge];
    int d = (int)ei[e + edge];
    float nrm = dis[s] * dis[d];
    unsafeAtomicAdd(out + d * 2 + k, nrm * T3[s * 2 + k]);
}

// ---------------- launch ----------------

extern "C" void kernel_launch(void* const* d_in, const int* in_sizes, int n_in,
                              void* d_out, int out_size, void* d_ws, size_t ws_size,
                              hipStream_t stream) {
    (void)in_sizes; (void)n_in; (void)out_size; (void)ws_size;

    const float*     x  = (const float*)d_in[0];
    const long long* ei = (const long long*)d_in[1];   // int64, shape (2, E)
    const float* Wl[3] = {(const float*)d_in[2], (const float*)d_in[4], (const float*)d_in[6]};
    const float* bl[3] = {(const float*)d_in[3], (const float*)d_in[5], (const float*)d_in[7]};
    const float* W3 = (const float*)d_in[8];
    const float* b3 = (const float*)d_in[9];
    float* out = (float*)d_out;

    // workspace layout
    float*    T    = (float*)d_ws;                     // N*256
    float*    Agg  = T + (size_t)NNODES * FDIM;        // N*256
    float*    dis  = Agg + (size_t)NNODES * FDIM;      // N
    unsigned* deg  = (unsigned*)(dis + NNODES);        // N
    float*    T3   = (float*)(deg + NNODES);           // N*2

    const int B = 256;

    // degree + dis = deg^-1/2 (deg >= 1 from self-loop)
    deg_init_kernel<<<(NNODES + B - 1) / B, B, 0, stream>>>(deg, NNODES);
    deg_edge_kernel<<<(NEDGES + B - 1) / B, B, 0, stream>>>(ei, deg, NEDGES);
    dis_kernel<<<(NNODES + B - 1) / B, B, 0, stream>>>(deg, dis, NNODES);

    // GEMM grid: 3125 row-tiles * 16 col-tiles = 50000 waves = 6250 blocks of 8 waves
    const int gemmBlocks = (NNODES / 16) * 16 / (B / 32);  // 6250
    const int aggTotal   = NNODES * FDIM;
    const long long edgeThreads = (long long)NEDGES * 64;
    const int edgeBlocks = (int)((edgeThreads + B - 1) / B);

    for (int l = 0; l < 3; ++l) {
        const float* Hin = (l == 0) ? x : Agg;
        if (l == 0)
            gemm256_wmma_kernel<0><<<gemmBlocks, B, 0, stream>>>(Hin, Wl[l], T);
        else
            gemm256_wmma_kernel<1><<<gemmBlocks, B, 0, stream>>>(Hin, Wl[l], T);
        agg_init_kernel<<<(aggTotal + B - 1) / B, B, 0, stream>>>(T, dis, bl[l], Agg, aggTotal);
        agg_edge_kernel<<<edgeBlocks, B, 0, stream>>>(ei, T, dis, Agg, NEDGES);
    }

    // final 256 -> 2 layer
    t3_kernel<<<(NNODES * 32 + B - 1) / B, B, 0, stream>>>(Agg, W3, T3, NNODES);
    out_init_kernel<<<(NNODES * 2 + B - 1) / B, B, 0, stream>>>(T3, dis, b3, out, NNODES * 2);
    out_edge_kernel<<<(NEDGES * 2 + B - 1) / B, B, 0, stream>>>(ei, T3, dis, out, NEDGES);
}